// PipeLine_84799834292727
// MI455X (gfx1250) — compile-verified
//
#include <hip/hip_runtime.h>
#include <math.h>

#define BB 8
#define NPT 1024
#define NITERS 20

typedef float v2f __attribute__((ext_vector_type(2)));
typedef float v8f __attribute__((ext_vector_type(8)));

// ---------------------------------------------------------------------------
// Per-point prep: 2-NN, ref_vec, sorted tri distances, |p|^2, ref.p, cross(ref,p)
// pd layout per point (12 floats): rx,ry,rz, t0,t1,t2, ccos, crx,cry,crz, n, pad
// grid (B, 2 structures), block 1024 (one thread per point)
// ---------------------------------------------------------------------------
__global__ __launch_bounds__(1024) void prep_kernel(const float* __restrict__ psrc,
                                                    const float* __restrict__ pref,
                                                    float* __restrict__ pd) {
  const int b = blockIdx.x;
  const int s = blockIdx.y;
  const float* P = (s == 0 ? psrc : pref) + (size_t)b * NPT * 3;
  __shared__ float sx[NPT], sy[NPT], sz[NPT];
  const int i = threadIdx.x;
  const float x = P[i * 3 + 0], y = P[i * 3 + 1], z = P[i * 3 + 2];
  sx[i] = x; sy[i] = y; sz[i] = z;
  __syncthreads();

  float b1 = 3.4e38f, b2 = 3.4e38f;
  int i1 = 0, i2 = 0;
  for (int j = 0; j < NPT; ++j) {
    if (j == i) continue;
    const float dx = sx[j] - x, dy = sy[j] - y, dz = sz[j] - z;
    const float d2 = dx * dx + dy * dy + dz * dz;
    if (d2 < b1)      { b2 = b1; i2 = i1; b1 = d2; i1 = j; }
    else if (d2 < b2) { b2 = d2; i2 = j; }
  }
  const float p1x = sx[i1], p1y = sy[i1], p1z = sz[i1];
  const float p2x = sx[i2], p2y = sy[i2], p2z = sz[i2];
  const float rx = p1x + p2x - x, ry = p1y + p2y - y, rz = p1z + p2z - z;

  const float d01 = sqrtf(b1);
  const float d02 = sqrtf(b2);
  const float ex = p1x - p2x, ey = p1y - p2y, ez = p1z - p2z;
  const float d12 = sqrtf(ex * ex + ey * ey + ez * ez);
  float a0 = d01, a1 = d12, a2 = d02, t;
  if (a0 > a1) { t = a0; a0 = a1; a1 = t; }
  if (a1 > a2) { t = a1; a1 = a2; a2 = t; }
  if (a0 > a1) { t = a0; a0 = a1; a1 = t; }

  const float n2   = x * x + y * y + z * z;
  const float ccos = rx * x + ry * y + rz * z;
  const float crx  = ry * z - rz * y;
  const float cry  = rz * x - rx * z;
  const float crz  = rx * y - ry * x;

  float4* o = (float4*)(pd + ((size_t)(s * BB + b) * NPT + i) * 12);
  o[0] = make_float4(rx, ry, rz, a0);
  o[1] = make_float4(a1, a2, ccos, crx);
  o[2] = make_float4(cry, crz, n2, 0.f);
}

// ---------------------------------------------------------------------------
// Sinkhorn row pass: r[b,i] = log(1 + sum_j exp(A[b,i,j] - c[j]))   (slack = 1)
// grid (N, B), block 256, float4 per thread (fully coalesced b128 loads)
// ---------------------------------------------------------------------------
__global__ __launch_bounds__(256) void sink_row_kernel(const float* __restrict__ A,
                                                       const float* __restrict__ cvec,
                                                       float* __restrict__ rvec, int useC) {
  const int i = blockIdx.x, b = blockIdx.y;
  const float4* row = (const float4*)(A + ((size_t)b * NPT + i) * NPT);
  const float4* cb  = (const float4*)(cvec + (size_t)b * NPT);
  float4 v = row[threadIdx.x];
  if (useC) {
    const float4 c4 = cb[threadIdx.x];
    v.x -= c4.x; v.y -= c4.y; v.z -= c4.z; v.w -= c4.w;
  }
  float m = fmaxf(fmaxf(fmaxf(v.x, v.y), fmaxf(v.z, v.w)), 0.0f);  // slack max
  __shared__ float red[256];
  red[threadIdx.x] = m; __syncthreads();
  for (int s2 = 128; s2 > 0; s2 >>= 1) {
    if (threadIdx.x < s2) red[threadIdx.x] = fmaxf(red[threadIdx.x], red[threadIdx.x + s2]);
    __syncthreads();
  }
  m = red[0]; __syncthreads();
  float ss = __expf(v.x - m) + __expf(v.y - m) + __expf(v.z - m) + __expf(v.w - m);
  red[threadIdx.x] = ss; __syncthreads();
  for (int s2 = 128; s2 > 0; s2 >>= 1) {
    if (threadIdx.x < s2) red[threadIdx.x] += red[threadIdx.x + s2];
    __syncthreads();
  }
  if (threadIdx.x == 0)
    rvec[(size_t)b * NPT + i] = m + __logf(red[0] + __expf(-m));
}

// ---------------------------------------------------------------------------
// Sinkhorn col pass: c[b,j] = log(1 + sum_i exp(A[b,i,j] - r[i]))
// grid (N/64, B), block 256 = 64 cols x 4 row-groups, online LSE, coalesced rows
// ---------------------------------------------------------------------------
__global__ __launch_bounds__(256) void sink_col_kernel(const float* __restrict__ A,
                                                       const float* __restrict__ rvec,
                                                       float* __restrict__ cvec) {
  const int b = blockIdx.y;
  const int j0 = blockIdx.x * 64;
  const int cidx = threadIdx.x & 63;
  const int g = threadIdx.x >> 6;
  __shared__ float rsh[NPT];
  for (int k = threadIdx.x; k < NPT; k += 256) rsh[k] = rvec[(size_t)b * NPT + k];
  __syncthreads();
  const float* Ab = A + (size_t)b * NPT * NPT;
  const int j = j0 + cidx;
  float m = -3.4e38f, ss = 0.f;
  for (int ii = 0; ii < 256; ++ii) {
    const int i = g * 256 + ii;
    const float tv = Ab[(size_t)i * NPT + j] - rsh[i];
    if (tv <= m) {
      ss += __expf(tv - m);
    } else {
      ss = ss * __expf(m - tv) + 1.0f;
      m = tv;
    }
  }
  __shared__ float pm[4][64], ps[4][64];
  pm[g][cidx] = m; ps[g][cidx] = ss;
  __syncthreads();
  if (threadIdx.x < 64) {
    float M = 0.0f;  // slack
#pragma unroll
    for (int q = 0; q < 4; ++q) M = fmaxf(M, pm[q][threadIdx.x]);
    float S = __expf(-M);  // slack term exp(0 - M)
#pragma unroll
    for (int q = 0; q < 4; ++q) S += ps[q][threadIdx.x] * __expf(pm[q][threadIdx.x] - M);
    cvec[(size_t)b * NPT + j0 + threadIdx.x] = M + __logf(S);
  }
}

// ---------------------------------------------------------------------------
// Fused pairwise kernel: one wave per 16x16 (i,j) tile.
// 5x V_WMMA_F32_16X16X4_F32 per structure: P.P^T, R.P^T, and 3 cross rows.
// Epilogue: dist/angle/tri_map + sinkhorn exp channel; LDS-staged b128 stores.
// grid (N/16, N/16, B), block 32
// ---------------------------------------------------------------------------
__global__ __launch_bounds__(32) void geo_pair_kernel(const float* __restrict__ psrc,
                                                      const float* __restrict__ pref,
                                                      const float* __restrict__ la,
                                                      const float* __restrict__ rvec,
                                                      const float* __restrict__ cvec,
                                                      const float* __restrict__ pd,
                                                      float* __restrict__ out) {
  const int b  = blockIdx.z;
  const int i0 = blockIdx.y * 16;
  const int j0 = blockIdx.x * 16;
  const int lane = threadIdx.x;
  const int m  = lane & 15;
  const int hi = lane >> 4;

  __shared__ __align__(16) float rowpd[16 * 12];
  __shared__ __align__(16) float colpd[16 * 12];
  __shared__ __align__(16) float stage[16 * 112];

  for (int s = 0; s < 2; ++s) {
    const float* P  = (s == 0 ? psrc : pref) + (size_t)b * NPT * 3;
    const float* PD = pd + (size_t)(s * BB + b) * NPT * 12;
    {
      const float4* prow = (const float4*)(PD + (size_t)i0 * 12);
      const float4* pcol = (const float4*)(PD + (size_t)j0 * 12);
      float4* srow = (float4*)rowpd;
      float4* scol = (float4*)colpd;
      for (int k = lane; k < 48; k += 32) {  // 192 floats = 48 float4
        srow[k] = prow[k];
        scol[k] = pcol[k];
      }
    }
    __syncthreads();

    const int ia = i0 + m;
    const float px = P[ia * 3 + 0], py = P[ia * 3 + 1], pz = P[ia * 3 + 2];
    const float rx = rowpd[m * 12 + 0], ry = rowpd[m * 12 + 1], rz = rowpd[m * 12 + 2];
    const int jb = j0 + m;
    const float qx = P[jb * 3 + 0], qy = P[jb * 3 + 1], qz = P[jb * 3 + 2];

    // A 16x4 layout: lanes0-15 K=0,1 ; lanes16-31 K=2,3 (K=3 zero pad)
    v2f aP  = { hi ? pz : px,   hi ? 0.f : py };
    v2f aR  = { hi ? rz : rx,   hi ? 0.f : ry };
    v2f aCx = { hi ? ry : 0.f,  hi ? 0.f : -rz };  // row (0,-rz,ry)
    v2f aCy = { hi ? -rx : rz,  0.f };             // row (rz,0,-rx)
    v2f aCz = { hi ? 0.f : -ry, hi ? 0.f : rx };   // row (-ry,rx,0)
    v2f bP  = { hi ? qz : qx,   hi ? 0.f : qy };   // B 4x16: col n=lane&15

    v8f zz = {0.f, 0.f, 0.f, 0.f, 0.f, 0.f, 0.f, 0.f};
    v8f dotT = __builtin_amdgcn_wmma_f32_16x16x4_f32(false, aP,  false, bP, (short)0, zz, false, false);
    v8f cosT = __builtin_amdgcn_wmma_f32_16x16x4_f32(false, aR,  false, bP, (short)0, zz, false, false);
    v8f cxT  = __builtin_amdgcn_wmma_f32_16x16x4_f32(false, aCx, false, bP, (short)0, zz, false, false);
    v8f cyT  = __builtin_amdgcn_wmma_f32_16x16x4_f32(false, aCy, false, bP, (short)0, zz, false, false);
    v8f czT  = __builtin_amdgcn_wmma_f32_16x16x4_f32(false, aCz, false, bP, (short)0, zz, false, false);

    const float nj  = colpd[m * 12 + 10];
    const float tj0 = colpd[m * 12 + 3], tj1 = colpd[m * 12 + 4], tj2 = colpd[m * 12 + 5];
#pragma unroll
    for (int v = 0; v < 8; ++v) {
      const int row = v + 8 * hi;          // C/D layout: lane group selects M+8
      const int gi = i0 + row, gj = j0 + m;
      const float ni  = rowpd[row * 12 + 10];
      const float cci = rowpd[row * 12 + 6];
      const float crx = rowpd[row * 12 + 7], cry = rowpd[row * 12 + 8], crz = rowpd[row * 12 + 9];
      const float ti0 = rowpd[row * 12 + 3], ti1 = rowpd[row * 12 + 4], ti2 = rowpd[row * 12 + 5];

      const float d2 = ni + nj - 2.f * dotT[v];
      float dist = sqrtf(fmaxf(d2, 0.f));
      const float cosv = cosT[v] - cci;
      const float sx2 = cxT[v] - crx, sy2 = cyT[v] - cry, sz2 = czT[v] - crz;
      const float sinv = sqrtf(sx2 * sx2 + sy2 * sy2 + sz2 * sz2);
      float ang = atan2f(sinv, cosv);
      float tm = fabsf(tj0 - ti0) + fabsf(tj1 - ti1) + fabsf(tj2 - ti2);
      if (gi == gj) { dist = 0.f; ang = 0.f; tm = 0.f; }  // exact diagonal

      stage[row * 112 + m * 7 + s * 3 + 0] = dist;
      stage[row * 112 + m * 7 + s * 3 + 1] = ang;
      stage[row * 112 + m * 7 + s * 3 + 2] = tm;
    }
    __syncthreads();
  }

  // channel 6: sinkhorn transport plan s = exp(A - r_i - c_j)
  {
    const float* lab = la + (size_t)b * NPT * NPT;
    const float* rb  = rvec + (size_t)b * NPT;
    const float cj = cvec[(size_t)b * NPT + j0 + m];
#pragma unroll
    for (int v = 0; v < 8; ++v) {
      const int row = v + 8 * hi;
      const int gi = i0 + row;
      stage[row * 112 + m * 7 + 6] = __expf(lab[(size_t)gi * NPT + (j0 + m)] - rb[gi] - cj);
    }
  }
  __syncthreads();

  // 16 rows x 112 floats: one global_store_b128 per row from 28 lanes
  for (int row = 0; row < 16; ++row) {
    float4* orow = (float4*)(out + (((size_t)b * NPT + (i0 + row)) * NPT + j0) * 7);
    if (lane < 28) orow[lane] = ((const float4*)(stage + row * 112))[lane];
  }
}

// ---------------------------------------------------------------------------
extern "C" void kernel_launch(void* const* d_in, const int* in_sizes, int n_in,
                              void* d_out, int out_size, void* d_ws, size_t ws_size,
                              hipStream_t stream) {
  (void)in_sizes; (void)n_in; (void)out_size; (void)ws_size;
  const float* psrc = (const float*)d_in[0];
  const float* pref = (const float*)d_in[1];
  const float* la   = (const float*)d_in[2];
  float* out = (float*)d_out;

  float* pd = (float*)d_ws;                               // 2*B*N*12 floats
  float* rv = pd + (size_t)2 * BB * NPT * 12;             // B*N
  float* cv = rv + (size_t)BB * NPT;                      // B*N

  prep_kernel<<<dim3(BB, 2), 1024, 0, stream>>>(psrc, pref, pd);

  for (int it = 0; it < NITERS; ++it) {
    sink_row_kernel<<<dim3(NPT, BB), 256, 0, stream>>>(la, cv, rv, it > 0 ? 1 : 0);
    sink_col_kernel<<<dim3(NPT / 64, BB), 256, 0, stream>>>(la, rv, cv);
  }

  geo_pair_kernel<<<dim3(NPT / 16, NPT / 16, BB), 32, 0, stream>>>(psrc, pref, la, rv, cv, pd, out);
}